// Detr3DCrossAttention_55937654063448
// MI455X (gfx1250) — compile-verified
//
#include <hip/hip_runtime.h>
#include <hip/hip_bf16.h>
#include <math.h>

#define QN   900
#define DN   256
#define NCAM 6
#define NLVL 4
#define NL   24   // NCAM*NLVL

typedef __attribute__((ext_vector_type(2))) float v2f;
typedef __attribute__((ext_vector_type(8))) float v8f;

// ---------------------------------------------------------------------------
// block-wide sum over 256 threads (shared-memory tree reduce)
// ---------------------------------------------------------------------------
__device__ __forceinline__ float block_sum_256(float v, float* sm) {
    int t = threadIdx.x;
    sm[t] = v;
    __syncthreads();
    #pragma unroll
    for (int s = 128; s > 0; s >>= 1) {
        if (t < s) sm[t] += sm[t + s];
        __syncthreads();
    }
    float r = sm[0];
    __syncthreads();
    return r;
}

// ---------------------------------------------------------------------------
// 1) Camera projection: per (cam, query) -> normalized grid xy + mask
// ---------------------------------------------------------------------------
__global__ void proj_kernel(const float* __restrict__ ref,
                            const float* __restrict__ l2i,
                            float* __restrict__ uv)
{
    int idx = blockIdx.x * blockDim.x + threadIdx.x;
    if (idx >= NCAM * QN) return;
    int n = idx / QN, q = idx % QN;
    float x = ref[q*3+0] * 102.4f - 51.2f;
    float y = ref[q*3+1] * 102.4f - 51.2f;
    float z = ref[q*3+2] * 8.0f   - 5.0f;
    const float* M = l2i + n * 16;
    float c0 = M[0]*x + M[1]*y + M[2]*z  + M[3];
    float c1 = M[4]*x + M[5]*y + M[6]*z  + M[7];
    float c2 = M[8]*x + M[9]*y + M[10]*z + M[11];
    bool infront = c2 > 1e-5f;
    float dz = fmaxf(c2, 1e-5f);
    float gx = (c0 / dz / 1600.f - 0.5f) * 2.f;
    float gy = (c1 / dz /  928.f - 0.5f) * 2.f;
    bool ok = infront && (gx > -1.f) && (gx < 1.f) && (gy > -1.f) && (gy < 1.f);
    float* o = uv + idx * 4;
    o[0] = gx; o[1] = gy; o[2] = ok ? 1.f : 0.f; o[3] = 0.f;
}

// ---------------------------------------------------------------------------
// Fragment loaders for the f32 16x16x4 WMMA k-chunks (K=32 -> 8 wmma steps).
// A is MxK row-major (lda=DN), B is KxN row-major (ldb).
// ---------------------------------------------------------------------------
__device__ __forceinline__ void load_chunk(const float* __restrict__ A,
                                           const float* __restrict__ B,
                                           int ldb, int row, int col,
                                           int kb, int half,
                                           v2f* __restrict__ a,
                                           v2f* __restrict__ b)
{
    #pragma unroll
    for (int i = 0; i < 8; ++i) {
        int k0 = kb + 4 * i + 2 * half;
        a[i] = *(const v2f*)(A + row * DN + k0);
        b[i].x = B[(k0    ) * ldb + col];
        b[i].y = B[(k0 + 1) * ldb + col];
    }
}

__device__ __forceinline__ void load_chunk_sum(const float* __restrict__ A0,
                                               const float* __restrict__ A1,
                                               const float* __restrict__ B,
                                               int ldb, int row, int col,
                                               int kb, int half,
                                               v2f* __restrict__ a,
                                               v2f* __restrict__ b)
{
    #pragma unroll
    for (int i = 0; i < 8; ++i) {
        int k0 = kb + 4 * i + 2 * half;
        v2f x = *(const v2f*)(A0 + row * DN + k0);
        v2f y = *(const v2f*)(A1 + row * DN + k0);
        a[i] = x + y;
        b[i].x = B[(k0    ) * ldb + col];
        b[i].y = B[(k0 + 1) * ldb + col];
    }
}

// ---------------------------------------------------------------------------
// 2) Attention weights: aw[q, n*4+l] = sigmoid((q+qp)·W_attn + b) * mask[n,q]
//    WMMA fp32 GEMM, double-buffered K chunks of 32
// ---------------------------------------------------------------------------
__global__ void attn_kernel(const float* __restrict__ query,
                            const float* __restrict__ qpos,
                            const float* __restrict__ Wa,
                            const float* __restrict__ ba,
                            const float* __restrict__ uv,
                            float* __restrict__ aw)
{
    int wid  = threadIdx.x >> 5;
    int lane = threadIdx.x & 31;
    int tile = blockIdx.x * 8 + wid;
    if (tile >= 57 * 2) return;               // wave-uniform exit
    int mt = tile >> 1, nt = tile & 1;
    int mbase = mt * 16, nbase = nt * 16;
    int half = lane >> 4, lm = lane & 15;
    int row = mbase + lm; if (row > QN - 1) row = QN - 1;
    int colc = nbase + lm; if (colc > NL - 1) colc = NL - 1;   // clamped for loads

    v2f af[2][8], bf[2][8];
    v8f acc = {};
    load_chunk_sum(query, qpos, Wa, NL, row, colc, 0, half, af[0], bf[0]);
    #pragma unroll
    for (int c = 0; c < 8; ++c) {
        if (c + 1 < 8)
            load_chunk_sum(query, qpos, Wa, NL, row, colc, (c + 1) * 32, half,
                           af[(c + 1) & 1], bf[(c + 1) & 1]);
        #pragma unroll
        for (int i = 0; i < 8; ++i)
            acc = __builtin_amdgcn_wmma_f32_16x16x4_f32(
                false, af[c & 1][i], false, bf[c & 1][i], (short)0, acc, false, false);
    }

    int scol = nbase + lm;
    if (scol < NL) {
        float bv  = ba[scol];
        int   cam = scol >> 2;
        #pragma unroll
        for (int r = 0; r < 8; ++r) {
            int rr = mbase + r + 8 * half;
            if (rr < QN) {
                float m = uv[(cam * QN + rr) * 4 + 2];
                float s = 1.f / (1.f + expf(-(acc[r] + bv)));
                aw[rr * NL + scol] = s * m;
            }
        }
    }
}

// ---------------------------------------------------------------------------
// 3) Bilinear sampling + fusion: fused[q,c] = sum_{n,l} aw * sample(feat)
// ---------------------------------------------------------------------------
__global__ void sample_fuse_kernel(const float* __restrict__ f0,
                                   const float* __restrict__ f1,
                                   const float* __restrict__ f2,
                                   const float* __restrict__ f3,
                                   const float* __restrict__ uv,
                                   const float* __restrict__ aw,
                                   float* __restrict__ fused)
{
    int q = blockIdx.x, c = threadIdx.x;
    const float* feats[4] = { f0, f1, f2, f3 };
    const int HH[4] = { 116, 58, 29, 15 };
    const int WW[4] = { 200, 100, 50, 25 };
    float acc = 0.f;
    for (int n = 0; n < NCAM; ++n) {
        const float* u = uv + (n * QN + q) * 4;
        float gx = u[0], gy = u[1], m = u[2];
        if (m == 0.f) continue;               // aw already masked to 0
        #pragma unroll
        for (int l = 0; l < NLVL; ++l) {
            float w = aw[q * NL + n * 4 + l];
            int H = HH[l], W = WW[l];
            float x = (gx + 1.f) * 0.5f * (float)W - 0.5f;
            float y = (gy + 1.f) * 0.5f * (float)H - 0.5f;
            float x0f = floorf(x), y0f = floorf(y);
            float wx1 = x - x0f, wy1 = y - y0f;
            float wx0 = 1.f - wx1, wy0 = 1.f - wy1;
            int x0 = (int)x0f, y0 = (int)y0f;
            const float* base = feats[l] + (size_t)(n * DN + c) * (size_t)(H * W);
            float s = 0.f;
            #pragma unroll
            for (int t = 0; t < 4; ++t) {
                int ix = x0 + (t & 1);
                int iy = y0 + (t >> 1);
                float tw = ((t & 1) ? wx1 : wx0) * ((t >> 1) ? wy1 : wy0);
                bool v = (ix >= 0) && (ix <= W - 1) && (iy >= 0) && (iy <= H - 1);
                int ixc = min(max(ix, 0), W - 1);
                int iyc = min(max(iy, 0), H - 1);
                s += base[iyc * W + ixc] * (v ? tw : 0.f);
            }
            acc += w * s;
        }
    }
    fused[q * DN + c] = acc;
}

// ---------------------------------------------------------------------------
// 4) PE branch stage 1: relu(LN(inverse_sigmoid(ref) @ W_pe1 + b_pe1))
// ---------------------------------------------------------------------------
__global__ void pe1_kernel(const float* __restrict__ ref,
                           const float* __restrict__ W1,
                           const float* __restrict__ b1,
                           const float* __restrict__ g1,
                           const float* __restrict__ be1,
                           float* __restrict__ t1)
{
    __shared__ float sm[256];
    int q = blockIdx.x, d = threadIdx.x;
    float h[3];
    #pragma unroll
    for (int i = 0; i < 3; ++i) {
        float v  = fminf(fmaxf(ref[q * 3 + i], 0.f), 1.f);
        float x1 = fmaxf(v, 1e-5f);
        float x2 = fmaxf(1.f - v, 1e-5f);
        h[i] = logf(x1 / x2);
    }
    float v = h[0] * W1[d] + h[1] * W1[DN + d] + h[2] * W1[2 * DN + d] + b1[d];
    float mu  = block_sum_256(v, sm) * (1.f / 256.f);
    float dv  = v - mu;
    float var = block_sum_256(dv * dv, sm) * (1.f / 256.f);
    float o   = dv * rsqrtf(var + 1e-5f) * g1[d] + be1[d];
    t1[q * DN + d] = fmaxf(o, 0.f);
}

// ---------------------------------------------------------------------------
// 5) Dual WMMA GEMM (M=900, N=256, K=256), double-buffered K chunks:
//    y==0: out = fused @ W_out + b_out       y==1: t2 = t1 @ W_pe2 + b_pe2
// ---------------------------------------------------------------------------
__global__ void gemm_dual_kernel(const float* __restrict__ fused,
                                 const float* __restrict__ Wout,
                                 const float* __restrict__ bout,
                                 const float* __restrict__ t1,
                                 const float* __restrict__ Wpe2,
                                 const float* __restrict__ bpe2,
                                 float* __restrict__ outA,
                                 float* __restrict__ t2)
{
    const float *A, *B, *bias;
    float* dst;
    if (blockIdx.y == 0) { A = fused; B = Wout; bias = bout; dst = outA; }
    else                 { A = t1;    B = Wpe2; bias = bpe2; dst = t2;   }

    int wid  = threadIdx.x >> 5;
    int lane = threadIdx.x & 31;
    int tile = blockIdx.x * 8 + wid;
    if (tile >= 57 * 16) return;              // wave-uniform
    int mt = tile >> 4, nt = tile & 15;
    int mbase = mt * 16, nbase = nt * 16;
    int half = lane >> 4, lm = lane & 15;
    int row = mbase + lm; if (row > QN - 1) row = QN - 1;
    int col = nbase + lm;

    v2f af[2][8], bf[2][8];
    v8f acc = {};
    load_chunk(A, B, DN, row, col, 0, half, af[0], bf[0]);
    #pragma unroll
    for (int c = 0; c < 8; ++c) {
        if (c + 1 < 8)
            load_chunk(A, B, DN, row, col, (c + 1) * 32, half,
                       af[(c + 1) & 1], bf[(c + 1) & 1]);
        #pragma unroll
        for (int i = 0; i < 8; ++i)
            acc = __builtin_amdgcn_wmma_f32_16x16x4_f32(
                false, af[c & 1][i], false, bf[c & 1][i], (short)0, acc, false, false);
    }

    float bv = bias[col];
    #pragma unroll
    for (int r = 0; r < 8; ++r) {
        int rr = mbase + r + 8 * half;
        if (rr < QN) dst[rr * DN + col] = acc[r] + bv;
    }
}

// ---------------------------------------------------------------------------
// 6) Finalize: out += relu(LN(t2))
// ---------------------------------------------------------------------------
__global__ void finalize_kernel(const float* __restrict__ t2,
                                const float* __restrict__ g2,
                                const float* __restrict__ be2,
                                float* __restrict__ out)
{
    __shared__ float sm[256];
    int q = blockIdx.x, d = threadIdx.x;
    float v   = t2[q * DN + d];
    float mu  = block_sum_256(v, sm) * (1.f / 256.f);
    float dv  = v - mu;
    float var = block_sum_256(dv * dv, sm) * (1.f / 256.f);
    float o   = dv * rsqrtf(var + 1e-5f) * g2[d] + be2[d];
    out[q * DN + d] += fmaxf(o, 0.f);
}

// ---------------------------------------------------------------------------
extern "C" void kernel_launch(void* const* d_in, const int* in_sizes, int n_in,
                              void* d_out, int out_size, void* d_ws, size_t ws_size,
                              hipStream_t stream)
{
    const float* query = (const float*)d_in[0];
    const float* qpos  = (const float*)d_in[1];
    const float* ref   = (const float*)d_in[2];
    const float* l2i   = (const float*)d_in[3];
    const float* f0    = (const float*)d_in[4];
    const float* f1    = (const float*)d_in[5];
    const float* f2    = (const float*)d_in[6];
    const float* f3    = (const float*)d_in[7];
    const float* Wa    = (const float*)d_in[8];
    const float* ba    = (const float*)d_in[9];
    const float* Wout  = (const float*)d_in[10];
    const float* bout  = (const float*)d_in[11];
    const float* Wp1   = (const float*)d_in[12];
    const float* bp1   = (const float*)d_in[13];
    const float* gp1   = (const float*)d_in[14];
    const float* bep1  = (const float*)d_in[15];
    const float* Wp2   = (const float*)d_in[16];
    const float* bp2   = (const float*)d_in[17];
    const float* gp2   = (const float*)d_in[18];
    const float* bep2  = (const float*)d_in[19];
    float* out = (float*)d_out;

    float* ws    = (float*)d_ws;
    float* uv    = ws;                 // 6*900*4      = 21600 floats
    float* aw    = uv + NCAM * QN * 4; // 900*24       = 21600 floats
    float* fused = aw + QN * NL;       // 900*256      = 230400 floats
    float* t1    = fused + QN * DN;    // 900*256
    float* t2    = t1 + QN * DN;       // 900*256  (total ~2.9 MB)

    proj_kernel<<<(NCAM * QN + 255) / 256, 256, 0, stream>>>(ref, l2i, uv);
    attn_kernel<<<15, 256, 0, stream>>>(query, qpos, Wa, ba, uv, aw);
    sample_fuse_kernel<<<QN, 256, 0, stream>>>(f0, f1, f2, f3, uv, aw, fused);
    pe1_kernel<<<QN, 256, 0, stream>>>(ref, Wp1, bp1, gp1, bep1, t1);
    dim3 g(114, 2);
    gemm_dual_kernel<<<g, 256, 0, stream>>>(fused, Wout, bout, t1, Wp2, bp2, out, t2);
    finalize_kernel<<<QN, 256, 0, stream>>>(t2, gp2, bep2, out);
}